// SetCriterion_31628139168167
// MI455X (gfx1250) — compile-verified
//
#include <hip/hip_runtime.h>
#include <math.h>

typedef float v2f __attribute__((ext_vector_type(2)));
typedef float v8f __attribute__((ext_vector_type(8)));

#define EPSF 1e-7f

__device__ __forceinline__ float softplusf(float x) {
    // numerically-stable log(1+exp(x))
    return fmaxf(x, 0.0f) + log1pf(expf(-fabsf(x)));
}

__device__ __forceinline__ float sigmoidf(float x) {
    return 1.0f / (1.0f + expf(-x));
}

// Full-wave (32 lane) sum reduction through the CDNA5 matrix unit.
// Place the lane partial in A-matrix K-slot 0 (A is 16x4 f32: lanes 0-15 hold
// K=0/1, lanes 16-31 hold K=2/3). With B = ones(4x16), every row of D equals
// rowsum_m = x[lane m] + x[lane m+16]. Each lane then holds 8 D values
// (M = 0..7 for lanes <16, M = 8..15 for lanes >=16); summing them and adding
// the opposite half via shfl_xor(16) yields the full 32-lane sum in all lanes.
// Must be called at a point where EXEC is all ones (uniform control flow).
__device__ __forceinline__ float wave_sum_wmma(float x) {
    v2f a; a[0] = x;    a[1] = 0.0f;
    v2f b; b[0] = 1.0f; b[1] = 1.0f;
    v8f c = {};
    v8f d = __builtin_amdgcn_wmma_f32_16x16x4_f32(
        /*neg_a=*/false, a, /*neg_b=*/false, b,
        /*c_mod=*/(short)0, c, /*reuse_a=*/false, /*reuse_b=*/false);
    float s = d[0] + d[1] + d[2] + d[3] + d[4] + d[5] + d[6] + d[7];
    s += __shfl_xor(s, 16, 32);
    return s;
}

__global__ void zero_ws_kernel(float* ws, int n) {
    int i = blockIdx.x * blockDim.x + threadIdx.x;
    int stride = gridDim.x * blockDim.x;
    for (; i < n; i += stride) ws[i] = 0.0f;
}

// One thread per target: gather ps, CIoU -> lbox, scatter tobj, class BCE -> lcls.
__global__ void target_kernel(const float* __restrict__ pi,
                              const int*   __restrict__ idx,
                              const float* __restrict__ tbox,
                              const float* __restrict__ anch,
                              const int*   __restrict__ lab,
                              float* __restrict__ tobj,
                              float* __restrict__ lbox_acc,
                              float* __restrict__ lcls_acc,
                              int N, int H) {
    int n = blockIdx.x * blockDim.x + threadIdx.x;
    float lbox_p = 0.0f;
    float lcls_p = 0.0f;

    if (n < N) {
        int b  = idx[0 * N + n];
        int a  = idx[1 * N + n];
        int gj = idx[2 * N + n];
        int gi = idx[3 * N + n];
        size_t cell = ((((size_t)b * 3 + a) * H + gj) * H + gi);
        const float* ps = pi + cell * 85;

        // pxy = sigmoid*2 - 0.5 ; pwh = (sigmoid*2)^2 * anchor
        float x1 = sigmoidf(ps[0]) * 2.0f - 0.5f;
        float y1 = sigmoidf(ps[1]) * 2.0f - 0.5f;
        float sw = sigmoidf(ps[2]) * 2.0f;
        float sh = sigmoidf(ps[3]) * 2.0f;
        float w1 = sw * sw * anch[n * 2 + 0];
        float h1 = sh * sh * anch[n * 2 + 1];

        float x2 = tbox[n * 4 + 0];
        float y2 = tbox[n * 4 + 1];
        float w2 = tbox[n * 4 + 2];
        float h2 = tbox[n * 4 + 3];

        // CIoU (forward value; alpha uses stop_gradient so plain value here)
        float b1x1 = x1 - w1 * 0.5f, b1x2 = x1 + w1 * 0.5f;
        float b1y1 = y1 - h1 * 0.5f, b1y2 = y1 + h1 * 0.5f;
        float b2x1 = x2 - w2 * 0.5f, b2x2 = x2 + w2 * 0.5f;
        float b2y1 = y2 - h2 * 0.5f, b2y2 = y2 + h2 * 0.5f;

        float iw = fmaxf(fminf(b1x2, b2x2) - fmaxf(b1x1, b2x1), 0.0f);
        float ih = fmaxf(fminf(b1y2, b2y2) - fmaxf(b1y1, b2y1), 0.0f);
        float inter = iw * ih;
        float uni = w1 * h1 + w2 * h2 - inter + EPSF;
        float iou = inter / uni;

        float cw = fmaxf(b1x2, b2x2) - fminf(b1x1, b2x1);
        float ch = fmaxf(b1y2, b2y2) - fminf(b1y1, b2y1);
        float c2 = cw * cw + ch * ch + EPSF;
        float dx = b2x1 + b2x2 - b1x1 - b1x2;
        float dy = b2y1 + b2y2 - b1y1 - b1y2;
        float rho2 = (dx * dx + dy * dy) * 0.25f;

        float da = atanf(w2 / (h2 + EPSF)) - atanf(w1 / (h1 + EPSF));
        float v = (4.0f / (float)(M_PI * M_PI)) * da * da;
        float alpha = v / (v - iou + (1.0f + EPSF));
        float ciou = iou - (rho2 / c2 + v * alpha);

        lbox_p = 1.0f - ciou;
        // scatter tobj (duplicate-index races are order-unspecified in the ref too)
        tobj[cell] = fmaxf(ciou, 0.0f);

        // one-hot class BCE: sum_c softplus(x_c) - x_label
        int l = lab[n];
        float cls = 0.0f;
        #pragma unroll 4
        for (int cidx = 0; cidx < 80; ++cidx) cls += softplusf(ps[5 + cidx]);
        cls -= ps[5 + l];
        lcls_p = cls;
    }

    // uniform point: all 32 lanes active
    float lbox_s = wave_sum_wmma(lbox_p);
    float lcls_s = wave_sum_wmma(lcls_p);
    if ((threadIdx.x & 31) == 0) {
        atomicAdd(lbox_acc, lbox_s);
        atomicAdd(lcls_acc, lcls_s);
    }
}

// Dense objectness BCE over the whole grid: sum( softplus(x) - t*x ).
__global__ void obj_kernel(const float* __restrict__ pi,
                           const float* __restrict__ tobj,
                           float* __restrict__ lobj_acc,
                           int M) {
    int i = blockIdx.x * blockDim.x + threadIdx.x;
    int stride = gridDim.x * blockDim.x;
    float acc = 0.0f;
    for (; i < M; i += stride) {
        // hint the next strided line into cache (speculative, OOB is dropped)
        __builtin_prefetch(pi + (size_t)(i + stride) * 85 + 4, 0, 3);
        float x = pi[(size_t)i * 85 + 4];
        float t = tobj[i];
        acc += softplusf(x) - t * x;
    }
    float s = wave_sum_wmma(acc);
    if ((threadIdx.x & 31) == 0) atomicAdd(lobj_acc, s);
}

__global__ void finalize_kernel(const float* __restrict__ acc,
                                float* __restrict__ out,
                                float invN0, float invN1, float invN2,
                                float invM0, float invM1, float invM2) {
    if (blockIdx.x == 0 && threadIdx.x == 0) {
        float lbox = (acc[0] * invN0 + acc[1] * invN1 + acc[2] * invN2) * 0.05f;
        float lcls = (acc[3] * invN0 + acc[4] * invN1 + acc[5] * invN2)
                     * (1.0f / 80.0f) * 0.5f;
        float lobj = (acc[6] * invM0 * 4.0f + acc[7] * invM1 * 1.0f
                      + acc[8] * invM2 * 0.4f) * 1.0f;
        float loss = lbox + lobj + lcls;
        out[0] = loss * 16.0f;  // loss * batch_size
        out[1] = lbox;
        out[2] = lobj;
        out[3] = lcls;
        out[4] = loss;
    }
}

extern "C" void kernel_launch(void* const* d_in, const int* in_sizes, int n_in,
                              void* d_out, int out_size, void* d_ws, size_t ws_size,
                              hipStream_t stream) {
    // setup_inputs() dict order: pi0, idx0, tbox0, anchors0, labels0, pi1, ...
    const float* pi[3];
    const int*   idx[3];
    const float* tbx[3];
    const float* anc[3];
    const int*   lab[3];
    int N[3];
    const int H[3] = {80, 40, 20};
    for (int i = 0; i < 3; ++i) {
        pi[i]  = (const float*)d_in[5 * i + 0];
        idx[i] = (const int*)  d_in[5 * i + 1];
        tbx[i] = (const float*)d_in[5 * i + 2];
        anc[i] = (const float*)d_in[5 * i + 3];
        lab[i] = (const int*)  d_in[5 * i + 4];
        N[i]   = in_sizes[5 * i + 4];   // labels element count
    }

    // workspace layout: [0..8] accumulators (lbox0-2, lcls0-2, lobj0-2),
    // padded to 16 floats, then the three dense tobj grids.
    float* acc = (float*)d_ws;
    float* tobj[3];
    int M[3];
    size_t off = 16;
    for (int i = 0; i < 3; ++i) {
        M[i] = 16 * 3 * H[i] * H[i];
        tobj[i] = acc + off;
        off += (size_t)M[i];
    }

    zero_ws_kernel<<<512, 256, 0, stream>>>(acc, (int)off);

    for (int i = 0; i < 3; ++i) {
        int blocks = (N[i] + 255) / 256;
        target_kernel<<<blocks, 256, 0, stream>>>(
            pi[i], idx[i], tbx[i], anc[i], lab[i],
            tobj[i], acc + i, acc + 3 + i, N[i], H[i]);
    }

    for (int i = 0; i < 3; ++i) {
        int blocks = (M[i] + 2047) / 2048;   // ~8 elements per thread
        if (blocks > 1024) blocks = 1024;
        obj_kernel<<<blocks, 256, 0, stream>>>(pi[i], tobj[i], acc + 6 + i, M[i]);
    }

    finalize_kernel<<<1, 32, 0, stream>>>(
        acc, (float*)d_out,
        1.0f / (float)N[0], 1.0f / (float)N[1], 1.0f / (float)N[2],
        1.0f / (float)M[0], 1.0f / (float)M[1], 1.0f / (float)M[2]);
}